// TrainablePCEN_47545287967308
// MI455X (gfx1250) — compile-verified
//
#include <hip/hip_runtime.h>

typedef __attribute__((ext_vector_type(2))) float v2f;
typedef __attribute__((ext_vector_type(4))) float v4f;
typedef __attribute__((ext_vector_type(8))) float v8f;

#define T_DIM 8192
#define SEG 512
#define LOOKBACK 64                      // (1-s)^64 ~ 2.6e-14 << f32 eps
#define NSEG (T_DIM / SEG)               // 16
#define WAVES_PER_BLOCK 8

// Fast pow for strictly-positive base: a^b = exp2(b * log2(a))
__device__ __forceinline__ float fast_pow_pos(float a, float b) {
  return __builtin_amdgcn_exp2f(b * __builtin_amdgcn_logf(a));
}

// PCEN via WMMA scan: one wave handles 16 rows x one time segment.
// EMA chunk (16 steps) computed as D = W(16x16) * X(16x16) + carry, using
// 4 chained v_wmma_f32_16x16x4_f32. Carry broadcast with ds_bpermute.
// PCEN tail uses native v_log_f32/v_exp_f32 (bases strictly positive) and
// contiguous b128 loads/stores in D-layout.
__global__ __launch_bounds__(256) void pcen_wmma_kernel(
    const float* __restrict__ x,
    const float* __restrict__ p_alpha,
    const float* __restrict__ p_delta,
    const float* __restrict__ p_r,
    const float* __restrict__ p_s,
    float* __restrict__ out,
    int n_rowgroups)
{
  const int lane = threadIdx.x & 31;
  const int widx = threadIdx.x >> 5;
  const int task = blockIdx.x * WAVES_PER_BLOCK + widx;
  const int rowgroup = task / NSEG;
  const int seg      = task % NSEG;
  if (rowgroup >= n_rowgroups) return;   // wave-uniform: EXEC stays all-ones

  const float alpha = fminf(fmaxf(p_alpha[0], 0.01f), 0.99f);
  const float delta = fabsf(p_delta[0]) + 1e-6f;
  const float r     = fminf(fmaxf(p_r[0], 0.01f), 1.0f);
  const float s     = p_s[0];
  const float decay = 1.0f - s;
  const float dr    = fast_pow_pos(delta, r);   // pow(delta_c, r_c)

  const int halfsel = lane >> 4;         // 0 | 1
  const int col     = lane & 15;         // column (row within rowgroup); A-row i

  // ---- Build A operands (constant): A[i][k] = (i>=k) ? s*decay^(i-k) : 0
  // A layout: lane holds row i = lane%16; VGPR v holds K = 4*kk + v + 2*(lane/16)
  v2f amat[4];
#pragma unroll
  for (int kk = 0; kk < 4; ++kk) {
#pragma unroll
    for (int v = 0; v < 2; ++v) {
      int k = 4 * kk + v + 2 * halfsel;
      float w = 0.0f;
      if (col >= k) {
        float pw = s;
        for (int t = 0; t < col - k; ++t) pw *= decay;
        w = pw;
      }
      amat[kk][v] = w;
    }
  }
  // Carry coefficients for C matrix: decay^(i+1), i = v + 8*halfsel
  float cdecay[8];
  {
    float pw = decay;
    for (int t = 0; t < 8 * halfsel; ++t) pw *= decay;
#pragma unroll
    for (int v = 0; v < 8; ++v) { cdecay[v] = pw; pw *= decay; }
  }

  const size_t row_idx = (size_t)rowgroup * 16 + (size_t)col;
  const float* __restrict__ xrow = x   + row_idx * (size_t)T_DIM;
  float*       __restrict__ orow = out + row_idx * (size_t)T_DIM;

  int t0, nchunks, warm_chunks;
  float m_in;
  if (seg == 0) {
    t0 = 0; nchunks = SEG / 16; warm_chunks = 0;
    m_in = xrow[0];                      // exact: (1-s)*x0 + s*x0 = x0 = M[0]
  } else {
    t0 = seg * SEG - LOOKBACK;
    nchunks = (SEG + LOOKBACK) / 16;
    warm_chunks = LOOKBACK / 16;
    m_in = 0.0f;                         // converges within lookback window
  }

  for (int c = 0; c < nchunks; ++c) {
    const int tbase = t0 + c * 16;

    // Prefetch next chunk's stream for this lane's row
    __builtin_prefetch(xrow + tbase + 16, 0, 1);

    // C matrix = carry contribution decay^(i+1) * m_in
    v8f acc;
#pragma unroll
    for (int v = 0; v < 8; ++v) acc[v] = cdecay[v] * m_in;

    // 4 chained WMMAs over K = 0..15
    // B layout: lane = column n, VGPR v holds K = 4*kk + v + 2*(lane/16)
#pragma unroll
    for (int kk = 0; kk < 4; ++kk) {
      const int tb = tbase + 4 * kk + 2 * halfsel;
      v2f b;
      b[0] = xrow[tb];
      b[1] = xrow[tb + 1];
      acc = __builtin_amdgcn_wmma_f32_16x16x4_f32(
          /*neg_a=*/false, amat[kk], /*neg_b=*/false, b,
          /*c_mod=*/(short)0, acc, /*reuse_a=*/false, /*reuse_b=*/false);
    }

    // Carry for next chunk: M[15] lives in D-VGPR7 of lanes 16..31 (col = lane-16)
    m_in = __int_as_float(
        __builtin_amdgcn_ds_bpermute((16 + col) << 2, __float_as_int(acc[7])));

    // PCEN elementwise + store (skip warm-up region).
    // D layout per lane: acc[v] <-> time tbase + 8*halfsel + v  (contiguous!)
    if (c >= warm_chunks) {
      const float* xp = xrow + tbase + 8 * halfsel;
      float*       op = orow + tbase + 8 * halfsel;
      const v4f x0 = *(const v4f*)(xp);
      const v4f x1 = *(const v4f*)(xp + 4);
      float xv8[8] = {x0[0], x0[1], x0[2], x0[3], x1[0], x1[1], x1[2], x1[3]};
      v4f o0, o1;
#pragma unroll
      for (int v = 0; v < 8; ++v) {
        const float M = acc[v];
        // bases strictly positive -> native log2/exp2 are safe
        const float smooth = fast_pow_pos(1e-6f + M, alpha);
        const float base   = xv8[v] * __builtin_amdgcn_rcpf(smooth + 1e-6f) + delta;
        const float p      = fast_pow_pos(base, r) - dr;
        if (v < 4) o0[v] = p; else o1[v - 4] = p;
      }
      *(v4f*)(op)     = o0;
      *(v4f*)(op + 4) = o1;
    }
  }
}

extern "C" void kernel_launch(void* const* d_in, const int* in_sizes, int n_in,
                              void* d_out, int out_size, void* d_ws, size_t ws_size,
                              hipStream_t stream) {
  const float* x       = (const float*)d_in[0];
  const float* p_alpha = (const float*)d_in[1];
  const float* p_delta = (const float*)d_in[2];
  const float* p_r     = (const float*)d_in[3];
  const float* p_s     = (const float*)d_in[4];
  float* out = (float*)d_out;

  const int n_rows = in_sizes[0] / T_DIM;          // 4096
  const int n_rowgroups = n_rows / 16;             // 256
  const int tasks = n_rowgroups * NSEG;            // 4096 waves
  const int blocks = (tasks + WAVES_PER_BLOCK - 1) / WAVES_PER_BLOCK;

  pcen_wmma_kernel<<<dim3(blocks), dim3(WAVES_PER_BLOCK * 32), 0, stream>>>(
      x, p_alpha, p_delta, p_r, p_s, out, n_rowgroups);
}